// Encoder_80582176407801
// MI455X (gfx1250) — compile-verified
//
#include <hip/hip_runtime.h>
#include <math.h>
#include <stdint.h>

typedef __attribute__((ext_vector_type(2))) float v2f;
typedef __attribute__((ext_vector_type(8))) float v8f;

#define HDIM        128
#define H_STRIDE    132   // h:  (4m + k) banks -> conflict-free A-fragment reads
#define W_STRIDE    136   // w2: disjoint bank halves for B-fragment reads
#define ROWS        128   // rows (edges/nodes) per group
#define NODES       100000
#define E_KEEP      1000000
#define E_TOTAL_LL  1100000LL

#if __has_builtin(__builtin_amdgcn_global_load_async_to_lds_b128) && \
    __has_builtin(__builtin_amdgcn_s_wait_asynccnt)
#define ASYNC_LDS_OK 1
// Builtin signature (from hipcc diagnostic): param0 = global (AS1) pointer to
// a generic 16-byte int vector, param1 = LDS (AS3) pointer, then imm offset
// and imm cpol.  Both pointers non-const, vector_size (not ext_vector_type).
typedef int v4i_vs __attribute__((vector_size(16)));
typedef __attribute__((address_space(1))) v4i_vs* gas_v4i;
typedef __attribute__((address_space(3))) v4i_vs* las_v4i;
#endif

// ---------------------------------------------------------------------------
// Stage w2 (128x128 f32, row-major) into LDS with W_STRIDE row padding.
// Uses gfx1250 async global->LDS copies when the toolchain exposes them.
// ---------------------------------------------------------------------------
__device__ __forceinline__ void stage_w2_lds(const float* __restrict__ w2,
                                             float* __restrict__ w2s, int tid)
{
#pragma unroll
    for (int i = 0; i < (HDIM * HDIM) / (4 * 256); ++i) {       // 16 iters
        const int idx  = (tid + i * 256) * 4;                   // float index
        const int row  = idx >> 7;
        const int col  = idx & (HDIM - 1);
#if defined(ASYNC_LDS_OK)
        __builtin_amdgcn_global_load_async_to_lds_b128(
            (gas_v4i)(w2 + idx),
            (las_v4i)(w2s + row * W_STRIDE + col),
            /*offset=*/0, /*cpol=*/0);
#else
        const float4 v = *(const float4*)(w2 + idx);
        *(float4*)(w2s + row * W_STRIDE + col) = v;
#endif
    }
#if defined(ASYNC_LDS_OK)
    __builtin_amdgcn_s_wait_asynccnt(0);
#endif
}

// ---------------------------------------------------------------------------
// Layer 2: out[rows] = h @ w2 + b2 via V_WMMA_F32_16X16X4_F32.
// Wave w owns a 16-wide N strip.  kc-outer / mt-inner with 8 resident C
// tiles (64 VGPRs) -> 8 independent WMMA dependency chains.
// ---------------------------------------------------------------------------
__device__ __forceinline__ void mlp_layer2_wmma(
    const float* __restrict__ h,        // LDS, ROWS x H_STRIDE
    const float* __restrict__ w2s,      // LDS, 128 x W_STRIDE
    float bias,                         // b2[ncol0 + nlane] (preloaded)
    float* __restrict__ out,            // n_rows x 128
    int row_base, int n_rows)
{
    const int lane  = threadIdx.x & 31;
    const int wave  = threadIdx.x >> 5;   // 0..7 -> N strip
    const int ncol0 = wave * 16;
    const int nlane = lane & 15;          // N col within strip / M row for A
    const int khalf = (lane >> 4) * 2;    // lanes 16-31 carry K = {4k+2,4k+3}

    v8f C[8];
#pragma unroll
    for (int mt = 0; mt < 8; ++mt)
#pragma unroll
        for (int i = 0; i < 8; ++i) C[mt][i] = bias;  // bias depends only on N

    const float* hbase = &h[nlane * H_STRIDE + khalf];
    const float* wbase = &w2s[khalf * W_STRIDE + ncol0 + nlane];

#pragma unroll
    for (int kc = 0; kc < 32; ++kc) {
        // B is 4x16 (KxN): VGPR0 = {K0 | K2}, VGPR1 = {K1 | K3}.
        v2f B;
        B.x = wbase[(kc * 4 + 0) * W_STRIDE];
        B.y = wbase[(kc * 4 + 1) * W_STRIDE];
#pragma unroll
        for (int mt = 0; mt < 8; ++mt) {
            // A is 16x4: both lane halves hold rows M=0..15; half picks K pair.
            v2f A;
            A.x = hbase[mt * 16 * H_STRIDE + kc * 4 + 0];
            A.y = hbase[mt * 16 * H_STRIDE + kc * 4 + 1];
            C[mt] = __builtin_amdgcn_wmma_f32_16x16x4_f32(
                        false, A, false, B, (short)0, C[mt], false, false);
        }
    }

    // D layout: row = vgpr + 8*(lane>=16), col = lane&15.  Streaming stores.
#pragma unroll
    for (int mt = 0; mt < 8; ++mt) {
#pragma unroll
        for (int v = 0; v < 8; ++v) {
            const int row = row_base + mt * 16 + v + ((lane >> 4) << 3);
            if (row < n_rows)
                __builtin_nontemporal_store(
                    C[mt][v], &out[(size_t)row * HDIM + ncol0 + nlane]);
        }
    }
}

// ---------------------------------------------------------------------------
// Edge embedding kernel (persistent, grid-stride over 128-edge groups).
// ---------------------------------------------------------------------------
__global__ __launch_bounds__(256) void edge_mlp_kernel(
    const float* __restrict__ pos,
    const long long* __restrict__ ei,     // 2 x 1,100,000 int64
    const float* __restrict__ ew1, const float* __restrict__ eb1,
    const float* __restrict__ ew2, const float* __restrict__ eb2,
    float* __restrict__ out)
{
    __shared__ float h[ROWS * H_STRIDE];
    __shared__ float w2s[HDIM * W_STRIDE];
    __shared__ float attr[ROWS * 4];

    const int tid  = threadIdx.x;
    const int lane = tid & 31;
    const float bias = eb2[(tid >> 5) * 16 + (lane & 15)];

    stage_w2_lds(ew2, w2s, tid);
    __syncthreads();

    const int ngroups = (E_KEEP + ROWS - 1) / ROWS;
    for (int g = blockIdx.x; g < ngroups; g += gridDim.x) {
        const int row_base = g * ROWS;

        if (tid < ROWS) {
            const int e  = row_base + tid;
            const int ec = (e < E_KEEP) ? e : 0;     // clamp; store is guarded
            const long long s = ei[ec];
            const long long d = ei[E_TOTAL_LL + ec];
            const float dx = pos[s * 3 + 0] - pos[d * 3 + 0];
            const float dy = pos[s * 3 + 1] - pos[d * 3 + 1];
            attr[tid * 4 + 0] = dx;
            attr[tid * 4 + 1] = dy;
            attr[tid * 4 + 2] = sqrtf(dx * dx + dy * dy);
        }
        __syncthreads();

        // Layer 1 (K=3) in VALU: h = relu(attr @ ew1 + eb1).
#pragma unroll 4
        for (int i = 0; i < (ROWS * HDIM) / 256; ++i) {
            const int idx = tid + i * 256;
            const int m = idx >> 7;
            const int c = idx & (HDIM - 1);
            float v = fmaf(attr[m * 4 + 0], ew1[0 * HDIM + c],
                      fmaf(attr[m * 4 + 1], ew1[1 * HDIM + c],
                      fmaf(attr[m * 4 + 2], ew1[2 * HDIM + c], eb1[c])));
            h[m * H_STRIDE + c] = v > 0.0f ? v : 0.0f;
        }
        __syncthreads();

        mlp_layer2_wmma(h, w2s, bias, out, row_base, E_KEEP);
        __syncthreads();   // h / attr reused next group
    }
}

// ---------------------------------------------------------------------------
// Node embedding kernel (persistent, grid-stride over 128-node groups).
// ---------------------------------------------------------------------------
__global__ __launch_bounds__(256) void node_mlp_kernel(
    const float* __restrict__ x,
    const float* __restrict__ nw1, const float* __restrict__ nb1,
    const float* __restrict__ nw2, const float* __restrict__ nb2,
    float* __restrict__ out)
{
    __shared__ float h[ROWS * H_STRIDE];
    __shared__ float w2s[HDIM * W_STRIDE];
    __shared__ float xs[ROWS * 8];

    const int tid  = threadIdx.x;
    const int lane = tid & 31;
    const float bias = nb2[(tid >> 5) * 16 + (lane & 15)];

    stage_w2_lds(nw2, w2s, tid);
    __syncthreads();

    const int ngroups = (NODES + ROWS - 1) / ROWS;
    for (int g = blockIdx.x; g < ngroups; g += gridDim.x) {
        const int row_base = g * ROWS;

        if (tid < ROWS) {
            const int r  = row_base + tid;
            const int rc = (r < NODES) ? r : 0;      // clamp; store is guarded
#pragma unroll
            for (int k = 0; k < 7; ++k)
                xs[tid * 8 + k] = x[(size_t)rc * 7 + k];
        }
        __syncthreads();

        // Layer 1 (K=7) in VALU.
#pragma unroll 4
        for (int i = 0; i < (ROWS * HDIM) / 256; ++i) {
            const int idx = tid + i * 256;
            const int m = idx >> 7;
            const int c = idx & (HDIM - 1);
            float v = nb1[c];
#pragma unroll
            for (int k = 0; k < 7; ++k)
                v = fmaf(xs[m * 8 + k], nw1[k * HDIM + c], v);
            h[m * H_STRIDE + c] = v > 0.0f ? v : 0.0f;
        }
        __syncthreads();

        mlp_layer2_wmma(h, w2s, bias, out, row_base, NODES);
        __syncthreads();
    }
}

// ---------------------------------------------------------------------------
// Launch.  Inputs (setup_inputs order):
//   0:x  1:pos  2:edge_index(int64)  3:nw1 4:nb1 5:nw2 6:nb2
//   7:ew1 8:eb1 9:ew2 10:eb2
// Output: node_embedding (1e5 x 128) then edge_embedding (1e6 x 128), f32.
// ---------------------------------------------------------------------------
extern "C" void kernel_launch(void* const* d_in, const int* in_sizes, int n_in,
                              void* d_out, int out_size, void* d_ws, size_t ws_size,
                              hipStream_t stream) {
    (void)in_sizes; (void)n_in; (void)out_size; (void)d_ws; (void)ws_size;

    const float*     x_p = (const float*)d_in[0];
    const float*     pos = (const float*)d_in[1];
    const long long* ei  = (const long long*)d_in[2];
    const float*     nw1 = (const float*)d_in[3];
    const float*     nb1 = (const float*)d_in[4];
    const float*     nw2 = (const float*)d_in[5];
    const float*     nb2 = (const float*)d_in[6];
    const float*     ew1 = (const float*)d_in[7];
    const float*     eb1 = (const float*)d_in[8];
    const float*     ew2 = (const float*)d_in[9];
    const float*     eb2 = (const float*)d_in[10];

    float* out_node = (float*)d_out;
    float* out_edge = out_node + (size_t)NODES * HDIM;

    node_mlp_kernel<<<256, 256, 0, stream>>>(
        x_p, nw1, nb1, nw2, nb2, out_node);
    edge_mlp_kernel<<<2048, 256, 0, stream>>>(
        pos, ei, ew1, eb1, ew2, eb2, out_edge);
}